// Network_14963666059445
// MI455X (gfx1250) — compile-verified
//
#include <hip/hip_runtime.h>
#include <math.h>

typedef float v2f __attribute__((ext_vector_type(2)));
typedef float v8f __attribute__((ext_vector_type(8)));

#define HD 16
#define NWIRE 1536
#define NFEAT 4

// ---------- helpers ----------
__device__ __forceinline__ unsigned int encOrd(float f) {
    unsigned int b = __float_as_uint(f);
    return (b & 0x80000000u) ? ~b : (b | 0x80000000u);
}
__device__ __forceinline__ float decOrd(unsigned int e) {
    unsigned int b = (e & 0x80000000u) ? (e ^ 0x80000000u) : ~e;
    return __uint_as_float(b);
}
__device__ __forceinline__ float lrelu02(float x) { return x > 0.f ? x : 0.2f * x; }

// ---------- pad W0 (13x16) -> (16x16), rows 13..15 = 0 ----------
__global__ void k_padW0(const float* __restrict__ W0, float* __restrict__ W0p) {
    int i = threadIdx.x;          // 256 threads
    int r = i >> 4;
    W0p[i] = (r < 13) ? W0[i] : 0.f;
}

// ---------- fill_window for graph g: h[n][0..15] (vectorized b128) ----------
__global__ void k_fill(const float* __restrict__ fw, const float* __restrict__ sw,
                       const float* __restrict__ tw, const float* __restrict__ areas,
                       const int* __restrict__ indices, float* __restrict__ h,
                       int N, int g) {
    int n = blockIdx.x * blockDim.x + threadIdx.x;
    if (n >= N) return;
    const size_t gbase = (size_t)g * NWIRE * NFEAT;
    const float4 f0 = *(const float4*)(fw + gbase + (size_t)indices[n * 3 + 0] * NFEAT);
    const float4 f1 = *(const float4*)(sw + gbase + (size_t)indices[n * 3 + 1] * NFEAT);
    const float4 f2 = *(const float4*)(tw + gbase + (size_t)indices[n * 3 + 2] * NFEAT);
    float4* hp = (float4*)(h + (size_t)n * HD);
    hp[0] = f0;
    hp[1] = f1;
    hp[2] = f2;
    hp[3] = make_float4(areas[n], 0.f, 0.f, 0.f);
}

// ---------- hw = h(Nx16) @ W(16x16) + bias, via V_WMMA_F32_16X16X4_F32 ----------
__global__ void k_gemm_wmma(const float* __restrict__ h, const float* __restrict__ W,
                            const float* __restrict__ bias, float* __restrict__ hw, int N) {
    const int lane = threadIdx.x & 31;
    const int wave = threadIdx.x >> 5;
    const int tile = blockIdx.x * 8 + wave;   // 8 waves per 256-thread block
    const int base = tile * 16;
    if (base >= N) return;                    // wave-uniform: EXEC stays all-ones
    const int col16 = lane & 15;              // N-column for B/C/D; M-row for A
    const int half  = lane >> 4;
    int arow = base + col16;                  // A-matrix row for this lane
    if (arow >= N) arow = N - 1;              // clamp (keeps EXEC full)
    v8f acc = {};
#pragma unroll
    for (int k = 0; k < 4; ++k) {
        const int kc = 4 * k + 2 * half;      // K index pair handled by this lane-half
        v2f a;                                // A 16x4 f32: VGPR0=K even, VGPR1=K odd
        a.x = h[(size_t)arow * HD + kc];
        a.y = h[(size_t)arow * HD + kc + 1];
        v2f b;                                // B 4x16 f32: row K striped across lanes
        b.x = W[kc * HD + col16];
        b.y = W[(kc + 1) * HD + col16];
        acc = __builtin_amdgcn_wmma_f32_16x16x4_f32(
                  false, a, false, b, (short)0, acc, false, false);
    }
    const float bi = bias[col16];
    if (base + 16 <= N) {                     // wave-uniform fast path: no guards
#pragma unroll
        for (int r = 0; r < 8; ++r) {         // D: VGPR r -> row (r + 8*half)
            hw[(size_t)(base + r + 8 * half) * HD + col16] = acc[r] + bi;
        }
    } else {                                  // tail tile only
#pragma unroll
        for (int r = 0; r < 8; ++r) {
            const int orow = base + r + 8 * half;
            if (orow < N) hw[(size_t)orow * HD + col16] = acc[r] + bi;
        }
    }
}

// ---------- per-node attention projections (b128 loads) ----------
__global__ void k_esed(const float* __restrict__ hw, const float* __restrict__ asrc,
                       const float* __restrict__ adst, float* __restrict__ es,
                       float* __restrict__ ed, int N) {
    int n = blockIdx.x * blockDim.x + threadIdx.x;
    if (n >= N) return;
    const float4* r = (const float4*)(hw + (size_t)n * HD);
    float s = 0.f, d = 0.f;
#pragma unroll
    for (int q = 0; q < 4; ++q) {
        const float4 v = r[q];
        s += v.x * asrc[q*4+0] + v.y * asrc[q*4+1] + v.z * asrc[q*4+2] + v.w * asrc[q*4+3];
        d += v.x * adst[q*4+0] + v.y * adst[q*4+1] + v.z * adst[q*4+2] + v.w * adst[q*4+3];
    }
    es[n] = s; ed[n] = d;
}

// ---------- per-layer init: m=-inf(encoded), denom=0, accum=0 ----------
__global__ void k_init(unsigned int* __restrict__ mEnc, float* __restrict__ denom,
                       float* __restrict__ accum, int N) {
    int j = blockIdx.x * blockDim.x + threadIdx.x;
    if (j >= N * HD) return;
    accum[j] = 0.f;
    if (j < N) { mEnc[j] = 0x007FFFFFu; /* encOrd(-inf) */ denom[j] = 0.f; }
}

// ---------- edge pass 1: segment max of logits (ordered-uint atomicMax) ----------
__global__ void k_edge_max(const int* __restrict__ src, const int* __restrict__ dst,
                           const float* __restrict__ es, const float* __restrict__ ed,
                           unsigned int* __restrict__ mEnc, int E) {
    int e = blockIdx.x * blockDim.x + threadIdx.x;
    if (e >= E) return;
    const int s = src[e], d = dst[e];
    const float lg = lrelu02(es[s] + ed[d]);
    atomicMax(&mEnc[d], encOrd(lg));
}

// ---------- edge pass 2: w=exp(logit-m[dst]); denom += w; cache w ----------
__global__ void k_edge_sum(const int* __restrict__ src, const int* __restrict__ dst,
                           const float* __restrict__ es, const float* __restrict__ ed,
                           const unsigned int* __restrict__ mEnc, float* __restrict__ denom,
                           float* __restrict__ wE, int E) {
    int e = blockIdx.x * blockDim.x + threadIdx.x;
    if (e >= E) return;
    const int s = src[e], d = dst[e];
    const float lg = lrelu02(es[s] + ed[d]);
    const float w = __expf(lg - decOrd(mEnc[d]));
    wE[e] = w;
    atomicAdd(&denom[d], w);
}

// ---------- edge pass 3: accum[dst] += alpha * hw[src], 4 lanes/edge, b128 loads ----
__global__ void k_edge_accum(const int* __restrict__ src, const int* __restrict__ dst,
                             const float* __restrict__ hw, const float* __restrict__ denom,
                             const float* __restrict__ wE, float* __restrict__ accum, int E) {
    int t = blockIdx.x * blockDim.x + threadIdx.x;
    if (t >= E * 4) return;
    const int e = t >> 2;
    const int q = t & 3;
    const int s = src[e], d = dst[e];
    const float alpha = wE[e] / fmaxf(denom[d], 1e-16f);
    const float4 hv = *(const float4*)(hw + (size_t)s * HD + q * 4);
    float* ap = accum + (size_t)d * HD + q * 4;
    atomicAdd(ap + 0, alpha * hv.x);
    atomicAdd(ap + 1, alpha * hv.y);
    atomicAdd(ap + 2, alpha * hv.z);
    atomicAdd(ap + 3, alpha * hv.w);
}

// ---------- h = elu(accum) ----------
__global__ void k_elu(const float* __restrict__ accum, float* __restrict__ h, int N) {
    int j = blockIdx.x * blockDim.x + threadIdx.x;
    if (j >= N * HD) return;
    const float x = accum[j];
    h[j] = x > 0.f ? x : (__expf(x) - 1.f);
}

// ---------- final MLP 16 -> 1 (b128 loads) ----------
__global__ void k_final(const float* __restrict__ h, const float* __restrict__ mlpW,
                        const float* __restrict__ mlpb, float* __restrict__ out, int N) {
    int n = blockIdx.x * blockDim.x + threadIdx.x;
    if (n >= N) return;
    const float4* r = (const float4*)(h + (size_t)n * HD);
    float acc = mlpb[0];
#pragma unroll
    for (int q = 0; q < 4; ++q) {
        const float4 v = r[q];
        acc += v.x * mlpW[q*4+0] + v.y * mlpW[q*4+1] + v.z * mlpW[q*4+2] + v.w * mlpW[q*4+3];
    }
    out[n] = acc;
}

static inline int cdiv(int a, int b) { return (a + b - 1) / b; }

extern "C" void kernel_launch(void* const* d_in, const int* in_sizes, int n_in,
                              void* d_out, int out_size, void* d_ws, size_t ws_size,
                              hipStream_t stream) {
    const float* fw     = (const float*)d_in[0];   // (1,T,NW,NF)
    const float* sw     = (const float*)d_in[1];
    const float* tw     = (const float*)d_in[2];
    const float* areas  = (const float*)d_in[3];   // (N)
    const int*   indices= (const int*)  d_in[4];   // (N,3)
    const int*   edge   = (const int*)  d_in[5];   // (2,E)
    const float* W0     = (const float*)d_in[6];   // (13,16)
    const float* Wl     = (const float*)d_in[7];   // (3,16,16)
    const float* a_src  = (const float*)d_in[8];   // (4,16)
    const float* a_dst  = (const float*)d_in[9];   // (4,16)
    const float* bvec   = (const float*)d_in[10];  // (4,16)
    const float* mlpW   = (const float*)d_in[11];  // (16,1)
    const float* mlpb   = (const float*)d_in[12];  // (1)

    const int N = in_sizes[3];
    const int E = in_sizes[5] / 2;
    const int T = in_sizes[0] / (NWIRE * NFEAT);
    const int* src = edge;
    const int* dst = edge + E;

    // workspace carve-out (floats; all sections 16B-aligned for N multiple of 4)
    float* ws = (float*)d_ws;
    size_t off = 0;
    float*        h      = ws + off; off += (size_t)N * HD;
    float*        hw     = ws + off; off += (size_t)N * HD;
    float*        accum  = ws + off; off += (size_t)N * HD;
    float*        es     = ws + off; off += (size_t)N;
    float*        ed     = ws + off; off += (size_t)N;
    unsigned int* mEnc   = (unsigned int*)(ws + off); off += (size_t)N;
    float*        denom  = ws + off; off += (size_t)N;
    float*        wE     = ws + off; off += (size_t)E;
    float*        W0p    = ws + off; off += 256;
    (void)ws_size; (void)n_in; (void)out_size;

    k_padW0<<<1, 256, 0, stream>>>(W0, W0p);

    const int tiles  = cdiv(N, 16);
    const int gemmBl = cdiv(tiles, 8);

    for (int g = 0; g < T; ++g) {
        k_fill<<<cdiv(N, 256), 256, 0, stream>>>(fw, sw, tw, areas, indices, h, N, g);
        for (int i = 0; i < 4; ++i) {
            const float* Wi = (i == 0) ? W0p : (Wl + (size_t)(i - 1) * 256);
            k_init<<<cdiv(N * HD, 256), 256, 0, stream>>>(mEnc, denom, accum, N);
            k_gemm_wmma<<<gemmBl, 256, 0, stream>>>(h, Wi, bvec + (size_t)i * HD, hw, N);
            k_esed<<<cdiv(N, 256), 256, 0, stream>>>(hw, a_src + (size_t)i * HD,
                                                     a_dst + (size_t)i * HD, es, ed, N);
            k_edge_max<<<cdiv(E, 256), 256, 0, stream>>>(src, dst, es, ed, mEnc, E);
            k_edge_sum<<<cdiv(E, 256), 256, 0, stream>>>(src, dst, es, ed, mEnc, denom, wE, E);
            k_edge_accum<<<cdiv(E * 4, 256), 256, 0, stream>>>(src, dst, hw, denom, wE, accum, E);
            k_elu<<<cdiv(N * HD, 256), 256, 0, stream>>>(accum, h, N);
        }
        k_final<<<cdiv(N, 256), 256, 0, stream>>>(h, mlpW, mlpb, (float*)d_out + (size_t)g * N, N);
    }
}